// QuantumLSTMCell_59279138619932
// MI455X (gfx1250) — compile-verified
//
#include <hip/hip_runtime.h>

typedef __attribute__((ext_vector_type(2))) float v2f;
typedef __attribute__((ext_vector_type(8))) float v8f;

#define BLOCK 256
#define WAVES 8
#define JSTRIDE 17  // odd stride -> conflict-free LDS rows

// CNOT(c, (c+1)%4) acting on a 4-bit basis index (wire 0 = MSB).
__device__ __forceinline__ int cnot_map(int k, int c) {
    int t = (c + 1) & 3;
    int cbit = (k >> (3 - c)) & 1;
    return k ^ (cbit << (3 - t));
}

// lane <-> lane^16 exchange within wave32 (ds_swizzle, group-of-32, xor=0x10, and=0x1f)
__device__ __forceinline__ float xor16(float v) {
    return __int_as_float(__builtin_amdgcn_ds_swizzle(__float_as_int(v), 0x401F));
}

__global__ __launch_bounds__(BLOCK) void qlstm_kernel(
    const float* __restrict__ x, const float* __restrict__ h_prev,
    const float* __restrict__ W_in, const float* __restrict__ b_in,
    const float* __restrict__ Up, const float* __restrict__ Ud,
    float* __restrict__ out, int Btot)
{
    __shared__ float trig[4][4][2];        // cos/sin(w/2) per layer,qubit
    __shared__ float Cmat[2][16][16];      // ping-pong circuit matrix
    __shared__ float ldsJ[WAVES][32][JSTRIDE];

    const int tid = threadIdx.x;

    // ---- Phase A: angles -> cos/sin (layer order: U0, U1, Ud1, Ud0) ----
    if (tid < 16) {
        int l = tid >> 2, q = tid & 3;
        float w = (l == 0) ? Up[q] : (l == 1) ? Up[4 + q]
                : (l == 2) ? Ud[4 + q] : Ud[q];
        trig[l][q][0] = cosf(0.5f * w);
        trig[l][q][1] = sinf(0.5f * w);
    }
    const int ii = tid >> 4, jj = tid & 15;
    Cmat[0][ii][jj] = (ii == jj) ? 1.0f : 0.0f;
    __syncthreads();

    // ---- Phase B: C = Layer3 @ Layer2 @ Layer1 @ Layer0 (each thread one element) ----
    int cur = 0;
    #pragma unroll
    for (int l = 0; l < 4; ++l) {
        // Layer[i][k] = U[pinv(i)][k]; pinv = g0∘g1∘g2∘g3
        int kk = ii;
        kk = cnot_map(kk, 3); kk = cnot_map(kk, 2);
        kk = cnot_map(kk, 1); kk = cnot_map(kk, 0);
        float sum = 0.0f;
        #pragma unroll
        for (int k = 0; k < 16; ++k) {
            float u = 1.0f;
            #pragma unroll
            for (int q = 0; q < 4; ++q) {
                int r  = (kk >> (3 - q)) & 1;
                int cb = (k  >> (3 - q)) & 1;
                float cc = trig[l][q][0], ss = trig[l][q][1];
                u *= (r == cb) ? cc : (r ? ss : -ss);   // RY = [[c,-s],[s,c]]
            }
            sum += u * Cmat[cur][k][jj];
        }
        Cmat[cur ^ 1][ii][jj] = sum;
        __syncthreads();
        cur ^= 1;
    }

    // ---- Phase C: load A fragments (A = C, 16x4 f32 per K-chunk) ----
    const int lane = tid & 31;
    const int wv   = tid >> 5;
    const int hh   = lane >> 4;      // lane half
    const int mn   = lane & 15;      // M for A, N for B/D
    v2f afrag[4];
    #pragma unroll
    for (int kc = 0; kc < 4; ++kc) {
        afrag[kc].x = Cmat[cur][mn][4 * kc + 2 * hh + 0];
        afrag[kc].y = Cmat[cur][mn][4 * kc + 2 * hh + 1];
    }
    float Wl[4], bl[4];
    #pragma unroll
    for (int j = 0; j < 4; ++j) { Wl[j] = W_in[j]; bl[j] = b_in[j]; }

    // ---- Phase D: streaming batch loop (32 elements per wave per iter) ----
    const int totalWaves = gridDim.x * WAVES;
    const int gw = blockIdx.x * WAVES + wv;
    const int stride = totalWaves * 32;

    for (int base = gw * 32; base < Btot; base += stride) {
        int e  = base + lane;
        int ec = min(e, Btot - 1);
        float xv = x[ec];
        float4 hp = ((const float4*)h_prev)[ec];

        // prefetch next iteration's h_prev
        int en = min(ec + stride, Btot - 1);
        __builtin_prefetch(&h_prev[4 * (size_t)en], 0, 1);

        // inp = proj / ||proj + 1e-12||
        float pj[4], tt, nn = 0.0f;
        #pragma unroll
        for (int j = 0; j < 4; ++j) {
            pj[j] = xv * Wl[j] + bl[j];
            tt = pj[j] + 1e-12f;
            nn += tt * tt;
        }
        float inv1 = 1.0f / sqrtf(nn);

        // joint = outer(inp, h_prev); normalize
        float jv[16];
        float hpv[4] = {hp.x, hp.y, hp.z, hp.w};
        float s2 = 0.0f;
        #pragma unroll
        for (int j = 0; j < 4; ++j) {
            float inj = pj[j] * inv1;
            #pragma unroll
            for (int k = 0; k < 4; ++k) {
                float t = inj * hpv[k];
                jv[4 * j + k] = t;
                s2 += t * t;
            }
        }
        float inv2 = 1.0f / sqrtf(s2);
        float* row = ldsJ[wv][lane];
        #pragma unroll
        for (int i2 = 0; i2 < 16; ++i2) row[i2] = jv[i2] * inv2;
        __builtin_amdgcn_wave_barrier();   // LDS is in-order within a wave

        #pragma unroll
        for (int half = 0; half < 2; ++half) {
            // B fragments: column N=mn is element (base + half*16 + mn)
            const float* jrow = ldsJ[wv][half * 16 + mn];
            v2f bfrag[4];
            #pragma unroll
            for (int kc = 0; kc < 4; ++kc) {
                bfrag[kc].x = jrow[4 * kc + 2 * hh + 0];
                bfrag[kc].y = jrow[4 * kc + 2 * hh + 1];
            }
            // psi[16x16] = C[16x16] * J[16x16] via 4x V_WMMA_F32_16X16X4_F32
            v8f acc = {};
            #pragma unroll
            for (int kc = 0; kc < 4; ++kc)
                acc = __builtin_amdgcn_wmma_f32_16x16x4_f32(
                    false, afrag[kc], false, bfrag[kc],
                    (short)0, acc, false, false);

            // lane holds psi rows {v + 8*hh} of its column; p = psi^2
            float p[8];
            #pragma unroll
            for (int v = 0; v < 8; ++v) p[v] = acc[v] * acc[v];
            // h_next[a] = sum over rows with row%4==a ; rows here are v+8hh -> a=v%4
            float ha0 = p[0] + p[4], ha1 = p[1] + p[5];
            float ha2 = p[2] + p[6], ha3 = p[3] + p[7];
            // <Z on wire1>: +1 for bit2(row)==0 -> v<4 in both halves
            float op = (p[0] + p[1] + p[2] + p[3]) - (p[4] + p[5] + p[6] + p[7]);
            // combine lane halves (rows 0-7 vs 8-15)
            ha0 += xor16(ha0); ha1 += xor16(ha1);
            ha2 += xor16(ha2); ha3 += xor16(ha3);
            op  += xor16(op);

            int eo = base + half * 16 + mn;
            if (hh == 0 && eo < Btot) {
                ((float4*)out)[eo] = make_float4(ha0, ha1, ha2, ha3); // h_next (B,4)
                out[4 * (size_t)Btot + eo] = op;                      // output (B,)
            }
        }
        __builtin_amdgcn_wave_barrier();
    }
}

extern "C" void kernel_launch(void* const* d_in, const int* in_sizes, int n_in,
                              void* d_out, int out_size, void* d_ws, size_t ws_size,
                              hipStream_t stream) {
    const float* x      = (const float*)d_in[0];
    const float* h_prev = (const float*)d_in[1];
    const float* W_in   = (const float*)d_in[2];
    const float* b_in   = (const float*)d_in[3];
    const float* Up     = (const float*)d_in[4];
    const float* Ud     = (const float*)d_in[5];
    float* out = (float*)d_out;

    int Btot = in_sizes[0];                 // x is (B,1)
    int tiles = (Btot + 32 * WAVES - 1) / (32 * WAVES);
    int blocks = 1024;                      // 8192 waves -> 4 sweeps at B=2^20
    if (blocks > tiles) blocks = (tiles > 0) ? tiles : 1;

    qlstm_kernel<<<blocks, BLOCK, 0, stream>>>(x, h_prev, W_in, b_in, Up, Ud, out, Btot);
}